// CenterLoss_120259084421
// MI455X (gfx1250) — compile-verified
//
#include <hip/hip_runtime.h>
#include <hip/hip_bf16.h>
#include <math.h>

typedef __attribute__((ext_vector_type(2))) float v2f;
typedef __attribute__((ext_vector_type(8))) float v8f;

#define NUM_CLASSES 1000
#define DIM 512

// ---------------------------------------------------------------------------
// Kernel 0: zero the per-class accumulator in workspace (must re-zero every
// call since the harness does not re-poison between graph replays).
// ---------------------------------------------------------------------------
__global__ void cl_zero_kernel(float* __restrict__ ws, int n) {
    int i = blockIdx.x * blockDim.x + threadIdx.x;
    if (i < n) ws[i] = 0.0f;
}

// ---------------------------------------------------------------------------
// Kernel 1: per-wave 16-sample tile. Accumulate diag(diff * diff^T) with
// V_WMMA_F32_16X16X4_F32 (full f32 precision), K-stepped by 4 over DIM=512.
//
// A-matrix 16x4 f32 layout (ISA 7.12.2): lane = m + 16*(k>=2), vgpr = k&1.
// B-matrix 4x16 layout is the same physical layout with N in place of M, so
// B == A register-for-register when computing diff * diff^T.
// Each lane therefore owns float2 { (m, kb+c), (m, kb+c+1) } with
// m = lane&15, c = 2*(lane>>4) -> one 8-byte load of x and of the gathered
// center row per WMMA step.
// ---------------------------------------------------------------------------
__global__ __launch_bounds__(256) void cl_sq_wmma_kernel(
    const float* __restrict__ x,
    const int*   __restrict__ labels,
    const float* __restrict__ centers,
    float*       __restrict__ per_class,
    int n_tiles)
{
    const int lane        = threadIdx.x & 31;
    const int wave_in_blk = threadIdx.x >> 5;
    const int tile        = blockIdx.x * (blockDim.x >> 5) + wave_in_blk;
    if (tile >= n_tiles) return;   // wave-uniform: EXEC stays all-1s for WMMA

    const int m      = lane & 15;                 // sample row within tile
    const int row    = tile * 16 + m;             // global sample index
    const int lab    = labels[row];               // lanes 16..31 mirror 0..15
    const int colsel = (lane >> 4) << 1;          // 0 for lanes<16, 2 otherwise

    const float* __restrict__ xrow = x       + (size_t)row * DIM + colsel;
    const float* __restrict__ crow = centers + (size_t)lab * DIM + colsel;

    v8f acc = {};
#pragma unroll 8
    for (int kb = 0; kb < DIM; kb += 4) {
        v2f xa = *(const v2f*)(xrow + kb);        // global_load_b64
        v2f ca = *(const v2f*)(crow + kb);        // L2-resident gather
        v2f d  = xa - ca;
        // D = A*B + C with A == B == diff fragment: diag(D) = per-sample sq.
        acc = __builtin_amdgcn_wmma_f32_16x16x4_f32(
            /*neg_a=*/false, d, /*neg_b=*/false, d,
            /*c_mod=*/(short)0, acc, /*reuse_a=*/false, /*reuse_b=*/false);
    }

    // Diagonal extraction via wave-private LDS slab.
    // C/D layout: VGPR r holds M=r (lanes 0-15, N=lane) and M=8+r
    // (lanes 16-31, N=lane-16).  (m,m) lives at vgpr m&7,
    // lane m (m<8) or lane m+16 (m>=8).
    __shared__ float red[8][32][8];
#pragma unroll
    for (int r = 0; r < 8; ++r) red[wave_in_blk][lane][r] = acc[r];
    // same-wave DS RAW: compiler inserts s_wait_dscnt; no barrier needed.

    if (lane < 16) {
        const int   r        = m & 7;
        const int   src_lane = (m < 8) ? m : (m + 16);
        const float sq       = red[wave_in_blk][src_lane][r];
        atomicAdd(&per_class[lab], sq);           // global_atomic_add_f32
    }
}

// ---------------------------------------------------------------------------
// Kernel 2: distance = sum(sqrt(per_class_sq)) / NUM_CLASSES, single block.
// ---------------------------------------------------------------------------
__global__ __launch_bounds__(1024) void cl_reduce_kernel(
    const float* __restrict__ per_class, float* __restrict__ out)
{
    __shared__ float s[1024];
    const int t = threadIdx.x;
    s[t] = (t < NUM_CLASSES) ? sqrtf(per_class[t]) : 0.0f;
    __syncthreads();
#pragma unroll
    for (int off = 512; off > 0; off >>= 1) {
        if (t < off) s[t] += s[t + off];
        __syncthreads();
    }
    if (t == 0) out[0] = s[0] / (float)NUM_CLASSES;
}

// ---------------------------------------------------------------------------
extern "C" void kernel_launch(void* const* d_in, const int* in_sizes, int n_in,
                              void* d_out, int out_size, void* d_ws, size_t ws_size,
                              hipStream_t stream) {
    const float* x       = (const float*)d_in[0];
    const int*   labels  = (const int*)  d_in[1];
    const float* centers = (const float*)d_in[2];
    float*       out     = (float*)d_out;
    float*       per_class = (float*)d_ws;        // NUM_CLASSES floats

    const int N       = in_sizes[1];              // 65536 samples
    const int n_tiles = N / 16;                   // 4096 wave-tiles

    cl_zero_kernel<<<(NUM_CLASSES + 255) / 256, 256, 0, stream>>>(per_class,
                                                                  NUM_CLASSES);

    const int waves_per_block = 256 / 32;         // 8 waves / block
    const int blocks = (n_tiles + waves_per_block - 1) / waves_per_block;
    cl_sq_wmma_kernel<<<blocks, 256, 0, stream>>>(x, labels, centers,
                                                  per_class, n_tiles);

    cl_reduce_kernel<<<1, 1024, 0, stream>>>(per_class, out);
}